// JNetwork_47356309406004
// MI455X (gfx1250) — compile-verified
//
#include <hip/hip_runtime.h>
#include <hip/hip_bf16.h>

typedef __attribute__((ext_vector_type(2))) float v2f;
typedef __attribute__((ext_vector_type(8))) float v8f;

#define N_SPECIES   2000
#define N_R1        80000
#define N_R2        10000
#define N_R3        10000
#define N_REACTIONS (N_R1 + N_R2 + N_R3)      /* 100000                     */
#define LINE_F      32                        /* floats per 128B line       */
#define N_LINES     (N_REACTIONS / LINE_F)    /* 3125 lines                 */
#define KSPLIT      25                        /* K-chunks -> grid.y         */
#define WAVES_PB    8                         /* 256 threads, wave32        */
#define GWAVES      (KSPLIT * WAVES_PB)       /* 200 waves per row tile     */
#define ROW_TILES   (N_SPECIES / 16)          /* 125                        */

// ---------------------------------------------------------------------------
// Phase 1: per-reaction rate coefficients * product of gathered abundances.
// ---------------------------------------------------------------------------
__global__ void __launch_bounds__(256) rates_kernel(
    const float* __restrict__ ab,          // [N_SPECIES]
    const float* __restrict__ temperature, // [1]
    const float* __restrict__ cr_rate,     // [1]
    const float* __restrict__ fuv_rate,    // [1]
    const float* __restrict__ av,          // [1]
    const int*   __restrict__ rm,          // [N_REACTIONS][2]
    const float* __restrict__ alpha,
    const float* __restrict__ beta,
    const float* __restrict__ gamma_,
    const float* __restrict__ zeta,
    const float* __restrict__ delta,
    const float* __restrict__ gamma_ph,
    float* __restrict__ rates)             // [N_REACTIONS] (workspace)
{
    int j = blockIdx.x * blockDim.x + threadIdx.x;
    if (j >= N_REACTIONS) return;

    float T = temperature[0];
    float r;
    if (j < N_R1) {
        // alpha * (T/300)^beta * exp(-gamma/T)
        r = alpha[j] * __expf(beta[j] * __logf(T * (1.0f / 300.0f)) - gamma_[j] / T);
    } else if (j < N_R1 + N_R2) {
        r = zeta[j - N_R1] * cr_rate[0];
    } else {
        int p = j - (N_R1 + N_R2);
        r = delta[p] * fuv_rate[0] * __expf(-gamma_ph[p] * av[0]);
    }

    int i0 = rm[2 * j + 0];
    int i1 = rm[2 * j + 1];
    float g0 = (i0 >= 0 && i0 < N_SPECIES) ? ab[i0] : 1.0f;
    float g1 = (i1 >= 0 && i1 < N_SPECIES) ? ab[i1] : 1.0f;
    rates[j] = r * g0 * g1;
}

// ---------------------------------------------------------------------------
// Phase 2: y_partial = incidence_tile @ rates via V_WMMA_F32_16X16X4_F32.
// A = 16x4 incidence tile; B = rates chunk replicated across all 16 columns,
// so every column of D carries the same dot-product partials; C accumulates.
// Each wave walks whole 128B lines in a grid-stride loop; 8 WMMAs per line.
// ---------------------------------------------------------------------------
__global__ void __launch_bounds__(256) gemv_wmma_kernel(
    const float* __restrict__ incidence,   // [N_SPECIES][N_REACTIONS] row-major
    const float* __restrict__ rates,       // [N_REACTIONS]
    float* __restrict__ partials)          // [KSPLIT][N_SPECIES]
{
    const int rowTile = blockIdx.x;            // 0..124
    const int kblk    = blockIdx.y;            // 0..24
    const int lane    = threadIdx.x & 31;
    const int wave    = threadIdx.x >> 5;      // 0..7
    const int row0    = rowTile * 16;
    const int m       = lane & 15;             // A-layout: both halves hold M=0..15
    const int koff    = (lane < 16) ? 0 : 2;   // VGPR0 = K0 (lo half) / K2 (hi half)
    const int g       = kblk * WAVES_PB + wave; // global wave id in row tile: 0..199

    const float* pa = incidence + (size_t)(row0 + m) * N_REACTIONS + g * LINE_F + koff;
    const float* pb = rates + g * LINE_F + koff;
    const int    stride = GWAVES * LINE_F;     // 6400 floats per outer step

    v8f c0 = {0.f, 0.f, 0.f, 0.f, 0.f, 0.f, 0.f, 0.f};
    v8f c1 = {0.f, 0.f, 0.f, 0.f, 0.f, 0.f, 0.f, 0.f};

    for (int l = g; l < N_LINES; l += GWAVES) {
        // Branchless in-bounds prefetch of the wave's next line (1 per 128B).
        const float* pf = pa + ((l + GWAVES < N_LINES) ? stride : 0);
        __builtin_prefetch(pf, 0, 3);

#pragma unroll
        for (int u = 0; u < 8; ++u) {
            v2f a = *(const v2f*)(pa + u * 4);   // global_load_b64, imm offset
            v2f b = *(const v2f*)(pb + u * 4);   // broadcast per half-wave
            if (u & 1)
                c1 = __builtin_amdgcn_wmma_f32_16x16x4_f32(false, a, false, b,
                                                           (short)0, c1, false, false);
            else
                c0 = __builtin_amdgcn_wmma_f32_16x16x4_f32(false, a, false, b,
                                                           (short)0, c0, false, false);
        }
        pa += stride;
        pb += stride;
    }

    v8f c = c0 + c1;

    // Column N=0 of D: lane 0 holds M=0..7 in c[0..7], lane 16 holds M=8..15.
    __shared__ float part[WAVES_PB][16];
    if (lane == 0) {
#pragma unroll
        for (int r = 0; r < 8; ++r) part[wave][r] = c[r];
    } else if (lane == 16) {
#pragma unroll
        for (int r = 0; r < 8; ++r) part[wave][8 + r] = c[r];
    }
    __syncthreads();

    // Fixed-order cross-wave reduction -> deterministic block partial.
    if (threadIdx.x < 16) {
        float s = 0.f;
#pragma unroll
        for (int w = 0; w < WAVES_PB; ++w) s += part[w][threadIdx.x];
        partials[(size_t)kblk * N_SPECIES + row0 + threadIdx.x] = s;
    }
}

// ---------------------------------------------------------------------------
// Phase 3: fixed-order reduction of the KSPLIT block partials (deterministic).
// ---------------------------------------------------------------------------
__global__ void __launch_bounds__(256) reduce_kernel(
    const float* __restrict__ partials,    // [KSPLIT][N_SPECIES]
    float* __restrict__ out)               // [N_SPECIES]
{
    int i = blockIdx.x * blockDim.x + threadIdx.x;
    if (i >= N_SPECIES) return;
    float s = 0.f;
#pragma unroll
    for (int b = 0; b < KSPLIT; ++b) s += partials[(size_t)b * N_SPECIES + i];
    out[i] = s;
}

// ---------------------------------------------------------------------------
extern "C" void kernel_launch(void* const* d_in, const int* in_sizes, int n_in,
                              void* d_out, int out_size, void* d_ws, size_t ws_size,
                              hipStream_t stream) {
    // setup_inputs() order:
    // 0 time, 1 abundances, 2 temperature, 3 cr_rate, 4 fuv_rate,
    // 5 visual_extinction, 6 incidence, 7 reactant_multipliers,
    // 8 alpha, 9 beta, 10 gamma, 11 zeta, 12 delta, 13 gamma_ph
    const float* ab        = (const float*)d_in[1];
    const float* temp      = (const float*)d_in[2];
    const float* cr        = (const float*)d_in[3];
    const float* fuv       = (const float*)d_in[4];
    const float* av        = (const float*)d_in[5];
    const float* incidence = (const float*)d_in[6];
    const int*   rm        = (const int*)d_in[7];
    const float* alpha     = (const float*)d_in[8];
    const float* beta      = (const float*)d_in[9];
    const float* gamma_    = (const float*)d_in[10];
    const float* zeta      = (const float*)d_in[11];
    const float* delta     = (const float*)d_in[12];
    const float* gph       = (const float*)d_in[13];

    float* rates    = (float*)d_ws;                 // 100000 floats
    float* partials = rates + N_REACTIONS;          // KSPLIT * 2000 floats

    rates_kernel<<<(N_REACTIONS + 255) / 256, 256, 0, stream>>>(
        ab, temp, cr, fuv, av, rm, alpha, beta, gamma_, zeta, delta, gph, rates);

    dim3 grid(ROW_TILES, KSPLIT);
    gemv_wmma_kernel<<<grid, 256, 0, stream>>>(incidence, rates, partials);

    reduce_kernel<<<(N_SPECIES + 255) / 256, 256, 0, stream>>>(
        partials, (float*)d_out);
}